// _AdapativeAggregation_25357486916159
// MI455X (gfx1250) — compile-verified
//
#include <hip/hip_runtime.h>
#include <hip/hip_bf16.h>
#include <cstdint>

// -------------------- types & helpers --------------------
typedef __attribute__((ext_vector_type(2))) float v2f;
typedef __attribute__((ext_vector_type(8))) float v8f;
typedef __attribute__((ext_vector_type(4))) unsigned int u32x4;
typedef __attribute__((ext_vector_type(8))) int i32x8;
typedef __attribute__((ext_vector_type(4))) int i32x4;

#define EPSF 1e-5f

__device__ __forceinline__ v8f wmma4(v2f a, v2f b, v8f c) {
  // D = A(16x4) * B(4x16) + C, fp32 end-to-end (matches fp32 reference)
  return __builtin_amdgcn_wmma_f32_16x16x4_f32(false, a, false, b, (short)0, c,
                                               false, false);
}

__device__ __forceinline__ int clampi(int v, int lo, int hi) {
  return v < lo ? lo : (v > hi ? hi : v);
}

// TDM: stage a (K rows x TW cols) fp32 tile of row-major B (ld = Nc) into LDS.
// Descriptor per CDNA5 ISA (2-D tensor, groups 2/3 zero).
// This toolchain exposes the 6-arg builtin: (g0, g1, g2, g3, g4, cpol).
__device__ __forceinline__ void tdm_stage_b(const float* Bt, float* lds_buf,
                                            int K, int Nc, int TW) {
  unsigned long long ga = (unsigned long long)(uintptr_t)Bt;   // global byte addr
  unsigned lds = (unsigned)(uintptr_t)lds_buf;                 // LDS byte offset
  u32x4 g0;
  g0.x = 1u;                                                   // count=1, user D#
  g0.y = lds;                                                  // lds_addr [63:32]
  g0.z = (unsigned)ga;                                         // global_addr lo
  g0.w = (unsigned)((ga >> 32) & 0x01FFFFFFull) | (2u << 30);  // addr hi | type=2
  i32x8 g1;
  g1[0] = 0x20000;                                   // wg_mask=0, data_size=4B
  g1[1] = (int)(((unsigned)Nc & 0xFFFFu) << 16);     // tensor_dim0 lo16
  g1[2] = (int)(((unsigned)K & 0xFFFFu) << 16);      // dim0 hi(0) | tensor_dim1 lo16
  g1[3] = (int)((unsigned)TW << 16);                 // dim1 hi(0) | tile_dim0
  g1[4] = K;                                         // tile_dim1 (tile_dim2 = 0)
  g1[5] = Nc;                                        // tensor_dim0_stride lo32
  g1[6] = 0;                                         // stride0 hi | stride1 lo
  g1[7] = 0;                                         // stride1 hi
  i32x4 g2 = {0, 0, 0, 0};
  i32x4 g3 = {0, 0, 0, 0};
  i32x8 g4 = {0, 0, 0, 0, 0, 0, 0, 0};
  __builtin_amdgcn_tensor_load_to_lds(g0, g1, g2, g3, g4, 0);
  __builtin_amdgcn_s_wait_tensorcnt(0);              // TDM done -> LDS valid
}

// -------------------- 1x1 conv (GEMM) with optional BN/relu/accumulate ------
// A: MxK row-major, B: KxN row-major, out: MxN. One wave = 16 x (NT*16) tile.
// B tile staged to LDS via TDM; K unrolled x4 -> 4*NT WMMA per iteration.
template <int NT>
__global__ __launch_bounds__(32)
void k_conv1x1(const float* __restrict__ A, const float* __restrict__ B,
               float* __restrict__ out,
               const float* __restrict__ bns, const float* __restrict__ bnb,
               const float* __restrict__ bnm, const float* __restrict__ bnv,
               int M, int K, int Nc, int act, int accum)
{
  __shared__ float bsh[64 * NT * 16];                 // K <= 64 at all call sites
  const int TW = NT * 16;
  const int ntile = Nc / TW;
  const int tm = blockIdx.x / ntile, tn = blockIdx.x % ntile;
  const int m0 = tm << 4, n0 = tn * TW;
  const int lane = threadIdx.x & 31;
  const int half = lane >> 4, l16 = lane & 15;

  tdm_stage_b(B + n0, bsh, K, Nc, TW);

  const float* Arow = A + (size_t)(m0 + l16) * K + half * 2;
  const v8f vzero = {};
  v8f acc[NT];
#pragma unroll
  for (int t = 0; t < NT; ++t) acc[t] = vzero;

  for (int k0 = 0; k0 < K; k0 += 16) {
    if (k0 + 16 < K) __builtin_prefetch(Arow + k0 + 16, 0, 1);
#pragma unroll
    for (int q = 0; q < 4; ++q) {
      v2f a;
      a.x = Arow[k0 + 4 * q];
      a.y = Arow[k0 + 4 * q + 1];
      const int ka = k0 + 4 * q + half * 2;
#pragma unroll
      for (int t = 0; t < NT; ++t) {
        v2f b;
        b.x = bsh[ka * TW + t * 16 + l16];
        b.y = bsh[(ka + 1) * TW + t * 16 + l16];
        acc[t] = wmma4(a, b, acc[t]);
      }
    }
  }
#pragma unroll
  for (int t = 0; t < NT; ++t) {
    const int col = n0 + t * 16 + l16;
    float mul = 1.f, add = 0.f;
    if (bns) { mul = bns[col] * rsqrtf(bnv[col] + EPSF); add = bnb[col] - bnm[col] * mul; }
#pragma unroll
    for (int r = 0; r < 8; ++r) {
      const int m = m0 + r + half * 8;
      float v = acc[t][r];
      if (bns) v = v * mul + add;
      if (act == 1) v = fmaxf(v, 0.f);
      const size_t idx = (size_t)m * Nc + col;
      if (accum) out[idx] += v; else out[idx] = v;
    }
  }
}

// -------------------- 1x1 conv + BN + residual(pad-mapped) + relu -----------
// Bottleneck tail: out[pad-mapped pixel] = relu(bn3(A@B) + resid[same pixel]).
template <int NT>
__global__ __launch_bounds__(32)
void k_conv1x1_res(const float* __restrict__ A, const float* __restrict__ B,
                   const float* __restrict__ resid, float* __restrict__ out,
                   const float* __restrict__ bns, const float* __restrict__ bnb,
                   const float* __restrict__ bnm, const float* __restrict__ bnv,
                   int Nb, int Ho, int Wo, int H, int W, int K, int Nc, int dpad)
{
  __shared__ float bsh[128 * NT * 16];                // K = wd <= 128
  const int TW = NT * 16;
  const int ntile = Nc / TW;
  const int tm = blockIdx.x / ntile, tn = blockIdx.x % ntile;
  const int m0 = tm << 4, n0 = tn * TW;
  const int lane = threadIdx.x & 31;
  const int half = lane >> 4, l16 = lane & 15;

  tdm_stage_b(B + n0, bsh, K, Nc, TW);

  const float* Arow = A + (size_t)(m0 + l16) * K + half * 2;
  const v8f vzero = {};
  v8f acc[NT];
#pragma unroll
  for (int t = 0; t < NT; ++t) acc[t] = vzero;

  for (int k0 = 0; k0 < K; k0 += 16) {
    if (k0 + 16 < K) __builtin_prefetch(Arow + k0 + 16, 0, 1);
#pragma unroll
    for (int q = 0; q < 4; ++q) {
      v2f a;
      a.x = Arow[k0 + 4 * q];
      a.y = Arow[k0 + 4 * q + 1];
      const int ka = k0 + 4 * q + half * 2;
#pragma unroll
      for (int t = 0; t < NT; ++t) {
        v2f b;
        b.x = bsh[ka * TW + t * 16 + l16];
        b.y = bsh[(ka + 1) * TW + t * 16 + l16];
        acc[t] = wmma4(a, b, acc[t]);
      }
    }
  }
  const int HoWo = Ho * Wo;
#pragma unroll
  for (int t = 0; t < NT; ++t) {
    const int col = n0 + t * 16 + l16;
    const float mul = bns[col] * rsqrtf(bnv[col] + EPSF);
    const float add = bnb[col] - bnm[col] * mul;
#pragma unroll
    for (int r = 0; r < 8; ++r) {
      const int m = m0 + r + half * 8;
      const int n = m / HoWo;
      const int rem = m % HoWo;
      const int ho = rem / Wo, wo = rem % Wo;
      const size_t pix = (((size_t)n * H + ho + dpad) * W + wo + dpad) * Nc + col;
      const float v = acc[t][r] * mul + add + resid[pix];
      out[pix] = fmaxf(v, 0.f);
    }
  }
}

// -------------------- generic 3x3 conv as implicit GEMM (WMMA) --------------
struct C3 {
  const float* x; const float* w; const float* bias;
  const float* bns; const float* bnb; const float* bnm; const float* bnv;
  float* out;
  int N, H, W, Cin, Ho, Wo, Cout;
  int stride, pad, dil, act; // act: 0 none, 1 relu, 2 leaky(0.2)
};

__global__ __launch_bounds__(32)
void k_conv3x3(C3 p)
{
  const int Kd = 9 * p.Cin;                  // Cin % 16 == 0 at all call sites
  const int ntile = (p.Cout + 15) >> 4;
  const int tm = blockIdx.x / ntile, tn = blockIdx.x % ntile;
  const int m0 = tm << 4, n0 = tn << 4;
  const int lane = threadIdx.x & 31;
  const int half = lane >> 4, l16 = lane & 15;
  const int m = m0 + l16;
  const int HoWo = p.Ho * p.Wo;
  const int n = m / HoWo;
  const int rem = m % HoWo;
  const int ho = rem / p.Wo, wo = rem % p.Wo;
  const int col = n0 + l16;
  const bool colok = (col < p.Cout);
  v8f acc = {};
  for (int k0 = 0; k0 < Kd; k0 += 16) {      // 16 K-values within one tap
    const int tap = k0 / p.Cin;
    const int ci0 = k0 - tap * p.Cin;
    const int iy = ho * p.stride - p.pad + (tap / 3) * p.dil;
    const int ix = wo * p.stride - p.pad + (tap % 3) * p.dil;
    const bool inb = (iy >= 0 && iy < p.H && ix >= 0 && ix < p.W);
    const float* px = p.x + (((size_t)n * p.H + iy) * p.W + ix) * p.Cin + ci0 + half * 2;
#pragma unroll
    for (int q = 0; q < 4; ++q) {
      v2f a = {}, b = {};
      if (inb) { a.x = px[4 * q]; a.y = px[4 * q + 1]; }
      const int ka = k0 + 4 * q + half * 2;
      if (colok) {
        b.x = p.w[(size_t)ka * p.Cout + col];
        b.y = p.w[(size_t)(ka + 1) * p.Cout + col];
      }
      acc = wmma4(a, b, acc);                // EXEC reconverged: full wave
    }
  }
  if (!colok) return;
  const float bia = p.bias ? p.bias[col] : 0.f;
  float mul = 1.f, add = 0.f;
  if (p.bns) { mul = p.bns[col] * rsqrtf(p.bnv[col] + EPSF); add = p.bnb[col] - p.bnm[col] * mul; }
#pragma unroll
  for (int r = 0; r < 8; ++r) {
    const int mm = m0 + r + half * 8;
    float v = acc[r] + bia;
    if (p.bns) v = v * mul + add;
    if (p.act == 1) v = fmaxf(v, 0.f);
    else if (p.act == 2) v = v >= 0.f ? v : 0.2f * v;
    p.out[(size_t)mm * p.Cout + col] = v;
  }
}

// -------------------- fused deformable-conv implicit GEMM (WMMA) ------------
// out[m,f] = relu(bn2( sum_{k,c} bilinear(x; off[m,k,g(c)])[c] * wdef[k,c,f] ))
// NT=2: the expensive sampled A fragment feeds two 16-col B tiles.
__global__ __launch_bounds__(32)
void k_deform(const float* __restrict__ x, const float* __restrict__ off,
              const float* __restrict__ wdef, float* __restrict__ out,
              const float* __restrict__ bns, const float* __restrict__ bnb,
              const float* __restrict__ bnm, const float* __restrict__ bnv,
              int Nb, int H, int W, int C, int Ho, int Wo, int dil)
{
  const int TW = 32;
  const int ntile = C / TW;                  // C (=wd) in {128,64,32}
  const int tm = blockIdx.x / ntile, tn = blockIdx.x % ntile;
  const int m0 = tm << 4, n0 = tn * TW;
  const int lane = threadIdx.x & 31;
  const int half = lane >> 4, l16 = lane & 15;
  const int m = m0 + l16;
  const int HoWo = Ho * Wo;
  const int n = m / HoWo;
  const int rem = m % HoWo;
  const int ho = rem / Wo, wo = rem % Wo;
  const int Cg = C >> 1;                     // G = 2 deform groups
  const int Kd = 9 * C;
  const float* xb = x + (size_t)n * H * W * C;
  const float* ob0 = off + (((size_t)n * Ho + ho) * Wo + wo) * 36; // (K=9,G=2,yx)
  const v8f vzero = {};
  v8f acc0 = vzero, acc1 = vzero;
  for (int k0 = 0; k0 < Kd; k0 += 4) {
    const int ka = k0 + half * 2;
    const int tap = ka / C;
    const int ci = ka - tap * C;
    const int g = (ci >= Cg) ? 1 : 0;
    const float oy = ob0[tap * 4 + g * 2 + 0];
    const float ox = ob0[tap * 4 + g * 2 + 1];
    const float py = (float)(ho + (tap / 3) * dil) + oy;
    const float px = (float)(wo + (tap % 3) * dil) + ox;
    const float fy = floorf(py), fx = floorf(px);
    const float wy = py - fy, wx = px - fx;
    const int y0 = clampi((int)fy, 0, H - 1);
    const int y1 = (y0 + 1 > H - 1) ? H - 1 : y0 + 1;
    const int x0 = clampi((int)fx, 0, W - 1);
    const int x1 = (x0 + 1 > W - 1) ? W - 1 : x0 + 1;
    const float* p00 = xb + ((size_t)y0 * W + x0) * C + ci;
    const float* p01 = xb + ((size_t)y0 * W + x1) * C + ci;
    const float* p10 = xb + ((size_t)y1 * W + x0) * C + ci;
    const float* p11 = xb + ((size_t)y1 * W + x1) * C + ci;
    const float w00 = (1.f - wy) * (1.f - wx), w01 = (1.f - wy) * wx;
    const float w10 = wy * (1.f - wx), w11 = wy * wx;
    v2f a;
    a.x = w00 * p00[0] + w01 * p01[0] + w10 * p10[0] + w11 * p11[0];
    a.y = w00 * p00[1] + w01 * p01[1] + w10 * p10[1] + w11 * p11[1];
    const float* wrow0 = wdef + (size_t)ka * C + n0 + l16;
    const float* wrow1 = wdef + (size_t)(ka + 1) * C + n0 + l16;
    v2f b0, b1;
    b0.x = wrow0[0];  b0.y = wrow1[0];
    b1.x = wrow0[16]; b1.y = wrow1[16];
    acc0 = wmma4(a, b0, acc0);
    acc1 = wmma4(a, b1, acc1);
  }
  {
    const int col = n0 + l16;
    const float mul = bns[col] * rsqrtf(bnv[col] + EPSF);
    const float add = bnb[col] - bnm[col] * mul;
#pragma unroll
    for (int r = 0; r < 8; ++r) {
      const int mm = m0 + r + half * 8;
      out[(size_t)mm * C + col] = fmaxf(acc0[r] * mul + add, 0.f);
    }
  }
  {
    const int col = n0 + 16 + l16;
    const float mul = bns[col] * rsqrtf(bnv[col] + EPSF);
    const float add = bnb[col] - bnm[col] * mul;
#pragma unroll
    for (int r = 0; r < 8; ++r) {
      const int mm = m0 + r + half * 8;
      out[(size_t)mm * C + col] = fmaxf(acc1[r] * mul + add, 0.f);
    }
  }
}

// -------------------- elementwise / resize / stats --------------------------
__global__ __launch_bounds__(256)
void k_copy(const float* __restrict__ a, float* __restrict__ b, size_t n) {
  const size_t i = (size_t)blockIdx.x * 256 + threadIdx.x;
  if (i < n) b[i] = a[i];
}

__global__ __launch_bounds__(256)
void k_copy_relu(const float* __restrict__ a, float* __restrict__ b, size_t n) {
  const size_t i = (size_t)blockIdx.x * 256 + threadIdx.x;
  if (i < n) b[i] = fmaxf(a[i], 0.f);
}

__global__ __launch_bounds__(256)
void k_resize(const float* __restrict__ in, float* __restrict__ out,
              int Nb, int Hi, int Wi, int Ho, int Wo, int C)
{
  const size_t total = (size_t)Nb * Ho * Wo * C;
  const size_t i = (size_t)blockIdx.x * 256 + threadIdx.x;
  if (i >= total) return;
  const int c = (int)(i % C);
  size_t r = i / C;
  const int xo = (int)(r % Wo); r /= Wo;
  const int yo = (int)(r % Ho);
  const int n = (int)(r / Ho);
  const float sy = (float)Hi / (float)Ho, sx = (float)Wi / (float)Wo;
  const float fyf = ((float)yo + 0.5f) * sy - 0.5f;
  const float fxf = ((float)xo + 0.5f) * sx - 0.5f;
  const float fy0 = floorf(fyf), fx0 = floorf(fxf);
  const float wy = fyf - fy0, wx = fxf - fx0;
  const int y0 = (int)fy0, x0 = (int)fx0;
  const int y0c = clampi(y0, 0, Hi - 1), y1c = clampi(y0 + 1, 0, Hi - 1);
  const int x0c = clampi(x0, 0, Wi - 1), x1c = clampi(x0 + 1, 0, Wi - 1);
  const float* b = in + (size_t)n * Hi * Wi * C + c;
  const float v00 = b[((size_t)y0c * Wi + x0c) * C];
  const float v01 = b[((size_t)y0c * Wi + x1c) * C];
  const float v10 = b[((size_t)y1c * Wi + x0c) * C];
  const float v11 = b[((size_t)y1c * Wi + x1c) * C];
  out[i] = (1.f - wy) * ((1.f - wx) * v00 + wx * v01)
         + wy * ((1.f - wx) * v10 + wx * v11);
}

__global__ __launch_bounds__(256)
void k_stats(const float* __restrict__ t, float* __restrict__ mean,
             float* __restrict__ var, int M, int C)
{
  __shared__ float ssum[256];
  __shared__ float ssq[256];
  const int c = blockIdx.x;
  float s = 0.f, q = 0.f;
  for (int i = threadIdx.x; i < M; i += 256) {
    const float v = t[(size_t)i * C + c];
    s += v; q += v * v;
  }
  ssum[threadIdx.x] = s; ssq[threadIdx.x] = q;
  __syncthreads();
  for (int st = 128; st > 0; st >>= 1) {
    if ((int)threadIdx.x < st) {
      ssum[threadIdx.x] += ssum[threadIdx.x + st];
      ssq[threadIdx.x]  += ssq[threadIdx.x + st];
    }
    __syncthreads();
  }
  if (threadIdx.x == 0) {
    const float mu = ssum[0] / (float)M;
    mean[c] = mu;
    var[c] = ssq[0] / (float)M - mu * mu;
  }
}

__global__ __launch_bounds__(256)
void k_bnb_add(const float* __restrict__ t, float* __restrict__ out,
               const float* __restrict__ mean, const float* __restrict__ var,
               const float* __restrict__ scale, const float* __restrict__ bias,
               size_t n, int C)
{
  const size_t i = (size_t)blockIdx.x * 256 + threadIdx.x;
  if (i < n) {
    const int c = (int)(i % C);
    out[i] += (t[i] - mean[c]) * rsqrtf(var[c] + EPSF) * scale[c] + bias[c];
  }
}

// -------------------- orchestration -----------------------------------------
extern "C" void kernel_launch(void* const* d_in, const int* in_sizes, int n_in,
                              void* d_out, int out_size, void* d_ws, size_t ws_size,
                              hipStream_t stream)
{
  (void)in_sizes; (void)n_in; (void)out_size; (void)ws_size;
  auto F = [&](int i) { return (const float*)d_in[i]; };
  float* ws = (float*)d_ws;
  float* outp = (float*)d_out;

  // workspace carve-up (floats), sized for scale 0
  float* OFFB = ws;                      // 2*92*188*36   = 1,245,312
  float* Y1   = OFFB + 1245312;          // 2*96*192*128  = 4,718,592
  float* Y2   = Y1 + 4718592;            // 2*92*188*128  = 4,427,776
  float* ISA0 = Y2 + 4427776;            // 2,359,296
  float* ISA1 = ISA0 + 2359296;          //   294,912
  float* ISA2 = ISA1 + 294912;           //    36,864
  float* T0   = ISA2 + 36864;            // 1,179,648
  float* T1   = T0 + 1179648;            //   589,824
  float* MEAN = T1 + 589824;             //        64
  float* VAR  = MEAN + 64;               //        64
  float* ISA[3] = {ISA0, ISA1, ISA2};

  const int CH[3] = {64, 32, 16};
  const int WD[3] = {128, 64, 32};
  const int HH[3] = {96, 48, 24};
  const int WWd[3] = {192, 96, 48};
  const int NB = 2, DILC = 2;

  // ---- per-scale ISA bottlenecks ----
  for (int s = 0; s < 3; ++s) {
    const int H = HH[s], W = WWd[s], c = CH[s], wd = WD[s];
    const int Ho = H - 2 * DILC, Wo = W - 2 * DILC;
    const int Mf = NB * H * W, Mi = NB * Ho * Wo;
    const int ib = 3 + s * 17;  // isa[s] params base (dict insertion order)
    const float* x = F(s);
    const float* w1 = F(ib + 0);
    const float *bn1s = F(ib + 1), *bn1b = F(ib + 2), *bn1m = F(ib + 3), *bn1v = F(ib + 4);
    const float *woff = F(ib + 5), *boff = F(ib + 6), *wdef = F(ib + 7);
    const float *bn2s = F(ib + 8), *bn2b = F(ib + 9), *bn2m = F(ib + 10), *bn2v = F(ib + 11);
    const float* w3 = F(ib + 12);
    const float *bn3s = F(ib + 13), *bn3b = F(ib + 14), *bn3m = F(ib + 15), *bn3v = F(ib + 16);

    // y1 = relu(bn1(x @ w1)); wd/32 N-supertiles
    k_conv1x1<2><<<dim3((Mf / 16) * (wd / 32)), dim3(32), 0, stream>>>(
        x, w1, Y1, bn1s, bn1b, bn1m, bn1v, Mf, c, wd, 1, 0);

    // offsets = conv3x3(dil=2, VALID)(y1) + b_off   (Cout=36, N-guarded)
    C3 pc = {Y1, woff, boff, nullptr, nullptr, nullptr, nullptr, OFFB,
             NB, H, W, wd, Ho, Wo, 36, 1, 0, DILC, 0};
    k_conv3x3<<<dim3((Mi / 16) * 3), dim3(32), 0, stream>>>(pc);

    // y2 = relu(bn2(deform_conv(y1, offsets, w_def)))
    k_deform<<<dim3((Mi / 16) * (wd / 32)), dim3(32), 0, stream>>>(
        Y1, OFFB, wdef, Y2, bn2s, bn2b, bn2m, bn2v, NB, H, W, wd, Ho, Wo, DILC);

    // isa = relu(x) on borders ...
    const size_t tot = (size_t)Mf * c;
    k_copy_relu<<<dim3((unsigned)((tot + 255) / 256)), dim3(256), 0, stream>>>(x, ISA[s], tot);
    // ... interior overwritten with relu(bn3(y2 @ w3) + x)
    if (c >= 32) {
      k_conv1x1_res<2><<<dim3((Mi / 16) * (c / 32)), dim3(32), 0, stream>>>(
          Y2, w3, x, ISA[s], bn3s, bn3b, bn3m, bn3v, NB, Ho, Wo, H, W, wd, c, DILC);
    } else {
      k_conv1x1_res<1><<<dim3((Mi / 16) * (c / 16)), dim3(32), 0, stream>>>(
          Y2, w3, x, ISA[s], bn3s, bn3b, bn3m, bn3v, NB, Ho, Wo, H, W, wd, c, DILC);
    }
  }

  // d_out starts as ISA (csa accumulators); ISA0..2 contiguous in ws
  k_copy<<<dim3((2691072 + 255) / 256), dim3(256), 0, stream>>>(ISA0, outp, 2691072);
  float* OUT0 = outp;
  float* OUT1 = outp + 2359296;
  float* OUT2 = outp + 2654208;

  // ---- CSA cross-scale terms ----
  // i=0, j=1: upsample isa1 -> 1x1 conv (32->64) accumulate
  {
    const size_t tot = (size_t)2 * 96 * 192 * 32;
    k_resize<<<dim3((unsigned)((tot + 255) / 256)), dim3(256), 0, stream>>>(
        ISA1, T0, 2, 48, 96, 96, 192, 32);
    k_conv1x1<2><<<dim3((36864 / 16) * 2), dim3(32), 0, stream>>>(
        T0, F(54), OUT0, nullptr, nullptr, nullptr, nullptr, 36864, 32, 64, 0, 1);
  }
  // i=0, j=2: upsample isa2 -> 1x1 conv (16->64) accumulate
  {
    const size_t tot = (size_t)2 * 96 * 192 * 16;
    k_resize<<<dim3((unsigned)((tot + 255) / 256)), dim3(256), 0, stream>>>(
        ISA2, T0, 2, 24, 48, 96, 192, 16);
    k_conv1x1<2><<<dim3((36864 / 16) * 2), dim3(32), 0, stream>>>(
        T0, F(55), OUT0, nullptr, nullptr, nullptr, nullptr, 36864, 16, 64, 0, 1);
  }
  // i=1, j=0: conv3x3 s2 (64->32) -> batch-stat BN -> accumulate
  {
    C3 pc = {ISA0, F(56), nullptr, nullptr, nullptr, nullptr, nullptr, T0,
             2, 96, 192, 64, 48, 96, 32, 2, 1, 1, 0};
    k_conv3x3<<<dim3((9216 / 16) * 2), dim3(32), 0, stream>>>(pc);
    k_stats<<<dim3(32), dim3(256), 0, stream>>>(T0, MEAN, VAR, 9216, 32);
    const size_t tot = (size_t)9216 * 32;
    k_bnb_add<<<dim3((unsigned)((tot + 255) / 256)), dim3(256), 0, stream>>>(
        T0, OUT1, MEAN, VAR, F(57), F(58), tot, 32);
  }
  // i=1, j=2: upsample isa2 -> 1x1 conv (16->32) accumulate
  {
    const size_t tot = (size_t)2 * 48 * 96 * 16;
    k_resize<<<dim3((unsigned)((tot + 255) / 256)), dim3(256), 0, stream>>>(
        ISA2, T0, 2, 24, 48, 48, 96, 16);
    k_conv1x1<2><<<dim3((9216 / 16) * 1), dim3(32), 0, stream>>>(
        T0, F(59), OUT1, nullptr, nullptr, nullptr, nullptr, 9216, 16, 32, 0, 1);
  }
  // i=2, j=0: conv3x3 s2 (64->64, BNrun+leaky) -> conv3x3 s2 (64->16) -> batch BN -> acc
  {
    C3 p1 = {ISA0, F(60), nullptr, F(61), F(62), F(63), F(64), T0,
             2, 96, 192, 64, 48, 96, 64, 2, 1, 1, 2};
    k_conv3x3<<<dim3((9216 / 16) * 4), dim3(32), 0, stream>>>(p1);
    C3 p2 = {T0, F(65), nullptr, nullptr, nullptr, nullptr, nullptr, T1,
             2, 48, 96, 64, 24, 48, 16, 2, 1, 1, 0};
    k_conv3x3<<<dim3((2304 / 16) * 1), dim3(32), 0, stream>>>(p2);
    k_stats<<<dim3(16), dim3(256), 0, stream>>>(T1, MEAN, VAR, 2304, 16);
    const size_t tot = (size_t)2304 * 16;
    k_bnb_add<<<dim3((unsigned)((tot + 255) / 256)), dim3(256), 0, stream>>>(
        T1, OUT2, MEAN, VAR, F(66), F(67), tot, 16);
  }
  // i=2, j=1: conv3x3 s2 (32->16) -> batch BN -> accumulate
  {
    C3 pc = {ISA1, F(68), nullptr, nullptr, nullptr, nullptr, nullptr, T0,
             2, 48, 96, 32, 24, 48, 16, 2, 1, 1, 0};
    k_conv3x3<<<dim3((2304 / 16) * 1), dim3(32), 0, stream>>>(pc);
    k_stats<<<dim3(16), dim3(256), 0, stream>>>(T0, MEAN, VAR, 2304, 16);
    const size_t tot = (size_t)2304 * 16;
    k_bnb_add<<<dim3((unsigned)((tot + 255) / 256)), dim3(256), 0, stream>>>(
        T0, OUT2, MEAN, VAR, F(69), F(70), tot, 16);
  }
}